// SparseGraphAttentionLayer_21569325761082
// MI455X (gfx1250) — compile-verified
//
#include <hip/hip_runtime.h>

#define N_NODES 100000
#define E_EDGES 1600000
#define IN_DIM  256
#define OUT_DIM 64
#define ALPHA   0.2f

typedef float v2f __attribute__((ext_vector_type(2)));
typedef float v8f __attribute__((ext_vector_type(8)));

#define LDS_STRIDE 260   // 256 + 4 pad: rows map to distinct LDS banks; 1040B rows stay 16B-aligned

// ---------------------------------------------------------------------------
// Kernel 1: Wh = X @ W via V_WMMA_F32_16X16X4_F32.
// Block = 128 threads (4 waves). Block owns a 16-row strip of X (staged in LDS,
// reused by all 4 waves); wave w owns output columns [16w, 16w+16).
// ---------------------------------------------------------------------------
__global__ __launch_bounds__(128) void gat_gemm_wmma(
    const float* __restrict__ X, const float* __restrict__ W,
    float* __restrict__ Wh)
{
    __shared__ __align__(16) float ldsA[16 * LDS_STRIDE];

    const int tid  = threadIdx.x;
    const int wave = tid >> 5;
    const int lane = tid & 31;
    const int m0   = blockIdx.x * 16;          // N_NODES % 16 == 0, no tail

    // Stage 16x256 f32 tile of X -> LDS with 128-bit coalesced loads.
    {
        const float4* X4 = (const float4*)(X + (size_t)m0 * IN_DIM);
#pragma unroll
        for (int it = 0; it < 8; ++it) {
            int f   = it * 128 + tid;          // float4 index in 16x64 grid
            int row = f >> 6;
            int c4  = f & 63;
            float4 v = X4[row * 64 + c4];
            *(float4*)&ldsA[row * LDS_STRIDE + c4 * 4] = v;
        }
    }
    __syncthreads();

    const int nBase = wave * 16;               // this wave's output columns
    const int mRow  = lane & 15;               // A: lane -> M row
    const int kHalf = (lane >> 4) * 2;         // A/B: upper half-wave holds K+2,K+3

    v8f c = {0.f, 0.f, 0.f, 0.f, 0.f, 0.f, 0.f, 0.f};

    const float* lA = &ldsA[mRow * LDS_STRIDE];
    const float* gB = W + nBase + (lane & 15); // B: lane -> N col, row-major W[k*64+n]

#pragma unroll 8
    for (int k0 = 0; k0 < IN_DIM; k0 += 4) {
        const int ka = k0 + kHalf;
        v2f a, b;
        a.x = lA[ka];                          // ds_load (bank-conflict-free via pad)
        a.y = lA[ka + 1];
        b.x = gB[(size_t)(ka)     * OUT_DIM];  // W is 64KB -> L2/L1 resident
        b.y = gB[(size_t)(ka + 1) * OUT_DIM];
        // (neg_a, A, neg_b, B, c_mod, C, reuse_a, reuse_b)
        c = __builtin_amdgcn_wmma_f32_16x16x4_f32(
                false, a, false, b, (short)0, c, false, false);
    }

    // C layout: VGPR r -> row (r + 8*(lane>=16)), col = lane&15.
    float* outp = Wh + (size_t)m0 * OUT_DIM + nBase + (lane & 15);
    const int rowOff = (lane >> 4) * 8;
#pragma unroll
    for (int r = 0; r < 8; ++r)
        outp[(size_t)(r + rowOff) * OUT_DIM] = c[r];
}

// ---------------------------------------------------------------------------
// Kernel 2: per-node attention scores + zero-init of accumulators.
// score_src[r] = Wh[r,:]·a[:64], score_dst[r] = Wh[r,:]·a[64:].
// Also zeroes out[r,:] and rowsum[r] (deterministic across graph replays).
// ---------------------------------------------------------------------------
__global__ void gat_scores_init(
    const float* __restrict__ Wh, const float* __restrict__ a,
    float* __restrict__ score_src, float* __restrict__ score_dst,
    float* __restrict__ rowsum, float* __restrict__ out)
{
    int r = blockIdx.x * blockDim.x + threadIdx.x;
    if (r >= N_NODES) return;

    const float4* wh4 = (const float4*)(Wh + (size_t)r * OUT_DIM);
    const float4* as4 = (const float4*)(a);
    const float4* ad4 = (const float4*)(a + OUT_DIM);
    float4* out4 = (float4*)(out + (size_t)r * OUT_DIM);

    float ss = 0.f, sd = 0.f;
#pragma unroll
    for (int j = 0; j < 16; ++j) {
        float4 w = wh4[j], s = as4[j], d = ad4[j];
        ss += w.x * s.x + w.y * s.y + w.z * s.z + w.w * s.w;
        sd += w.x * d.x + w.y * d.y + w.z * d.z + w.w * d.w;
        out4[j] = make_float4(0.f, 0.f, 0.f, 0.f);
    }
    score_src[r] = ss;
    score_dst[r] = sd;
    rowsum[r]    = 0.f;
}

// ---------------------------------------------------------------------------
// Kernel 3: per-edge weight + scatter-add. 16 lanes per edge, 4 cols per lane
// (64B-coalesced gather of Wh[dst]); f32 atomics land in L2 (Wh+out resident).
// ---------------------------------------------------------------------------
__global__ void gat_edge_scatter(
    const float* __restrict__ Wh, const int* __restrict__ edges,
    const float* __restrict__ score_src, const float* __restrict__ score_dst,
    float* __restrict__ rowsum, float* __restrict__ out)
{
    long long gid = (long long)blockIdx.x * blockDim.x + threadIdx.x;
    int e   = (int)(gid >> 4);
    int sub = (int)(gid & 15);
    if (e >= E_EDGES) return;

    int src = edges[e];
    int dst = edges[E_EDGES + e];

    float s  = score_src[src] + score_dst[dst];
    float lr = s > 0.f ? s : ALPHA * s;        // leaky_relu
    float w  = expf(-lr);

    if (sub == 0) atomicAdd(&rowsum[src], w);

    const float* whd = Wh + (size_t)dst * OUT_DIM;
    float*       o   = out + (size_t)src * OUT_DIM;
#pragma unroll
    for (int k = 0; k < 4; ++k) {
        int j = sub + k * 16;
        atomicAdd(&o[j], w * whd[j]);
    }
}

// ---------------------------------------------------------------------------
// Kernel 4: h_prime = h / rowsum, then ELU (alpha=1). In-place on d_out.
// ---------------------------------------------------------------------------
__global__ void gat_finalize(const float* __restrict__ rowsum, float* __restrict__ out)
{
    long long gid = (long long)blockIdx.x * blockDim.x + threadIdx.x;
    int r = (int)(gid >> 4);
    int q = (int)(gid & 15);
    if (r >= N_NODES) return;

    float inv = 1.f / rowsum[r];
    float4* o4 = (float4*)out + (size_t)r * 16 + q;
    float4 v = *o4;
    v.x *= inv; v.y *= inv; v.z *= inv; v.w *= inv;
    v.x = v.x > 0.f ? v.x : expm1f(v.x);
    v.y = v.y > 0.f ? v.y : expm1f(v.y);
    v.z = v.z > 0.f ? v.z : expm1f(v.z);
    v.w = v.w > 0.f ? v.w : expm1f(v.w);
    *o4 = v;
}

// ---------------------------------------------------------------------------
extern "C" void kernel_launch(void* const* d_in, const int* in_sizes, int n_in,
                              void* d_out, int out_size, void* d_ws, size_t ws_size,
                              hipStream_t stream)
{
    const float* X     = (const float*)d_in[0];   // N x 256
    const float* W     = (const float*)d_in[1];   // 256 x 64
    const float* a     = (const float*)d_in[2];   // 1 x 128
    const int*   edges = (const int*)d_in[3];     // 2 x E

    float* out = (float*)d_out;                   // N x 64

    // Workspace: Wh (N*64) | score_src (N) | score_dst (N) | rowsum (N)
    float* Wh        = (float*)d_ws;
    float* score_src = Wh + (size_t)N_NODES * OUT_DIM;
    float* score_dst = score_src + N_NODES;
    float* rowsum    = score_dst + N_NODES;

    gat_gemm_wmma<<<N_NODES / 16, 128, 0, stream>>>(X, W, Wh);

    gat_scores_init<<<(N_NODES + 255) / 256, 256, 0, stream>>>(
        Wh, a, score_src, score_dst, rowsum, out);

    long long edgeThreads = (long long)E_EDGES * 16;
    gat_edge_scatter<<<(int)((edgeThreads + 255) / 256), 256, 0, stream>>>(
        Wh, edges, score_src, score_dst, rowsum, out);

    long long finThreads = (long long)N_NODES * 16;
    gat_finalize<<<(int)((finThreads + 255) / 256), 256, 0, stream>>>(rowsum, out);
}